// GCN_48473000903499
// MI455X (gfx1250) — compile-verified
//
#include <hip/hip_runtime.h>
#include <hip/hip_bf16.h>

typedef __attribute__((ext_vector_type(2))) float v2f;
typedef __attribute__((ext_vector_type(8))) float v8f;
typedef __attribute__((ext_vector_type(4))) int   v4i;

#define DFEAT 64
#define NLAYERS 3
#define NOUT 16

// address-space-qualified int4 pointers for the async global->LDS builtin
#define ASG_V4I(p) ((__attribute__((address_space(1))) v4i*)(void*)(p))
#define ASL_V4I(p) ((__attribute__((address_space(3))) v4i*)(void*)(p))

// ---------------- degree / norm kernels ----------------

__global__ void gcn_init_deg(float* ns, float* nd, int n) {
    int i = blockIdx.x * blockDim.x + threadIdx.x;
    if (i < n) { ns[i] = 1.0f; nd[i] = 1.0f; }   // self loop contributes 1
}

__global__ void gcn_count_deg(float* ns, float* nd, const int* src, const int* dst, int e) {
    int i = blockIdx.x * blockDim.x + threadIdx.x;
    if (i < e) {
        atomicAdd(&ns[src[i]], 1.0f);
        atomicAdd(&nd[dst[i]], 1.0f);
    }
}

__global__ void gcn_finish_norm(float* ns, float* nd, int n) {
    int i = blockIdx.x * blockDim.x + threadIdx.x;
    if (i < n) {
        ns[i] = rsqrtf(ns[i]);
        nd[i] = rsqrtf(nd[i]);
    }
}

// ---------------- aggregation (self loop init + edge scatter) ----------------

__global__ void gcn_init_agg(float* agg, const float* x, const float* ns, long total) {
    long i = (long)blockIdx.x * blockDim.x + threadIdx.x;
    if (i < total) {
        long v = i >> 6;            // / DFEAT
        agg[i] = x[i] * ns[v];
    }
}

// 16 threads per edge, 4 floats each
__global__ void gcn_scatter(float* agg, const float* x, const float* ns,
                            const int* src, const int* dst, int e) {
    long t = (long)blockIdx.x * blockDim.x + threadIdx.x;
    int eidx = (int)(t >> 4);
    int part = (int)(t & 15);
    if (eidx < e) {
        int s = src[eidx];
        int d = dst[eidx];
        float w = ns[s];
        const float4 v = *(const float4*)(x + (size_t)s * DFEAT + part * 4);
        float* o = agg + (size_t)d * DFEAT + part * 4;
        atomicAdd(o + 0, v.x * w);
        atomicAdd(o + 1, v.y * w);
        atomicAdd(o + 2, v.z * w);
        atomicAdd(o + 3, v.w * w);
    }
}

// ---------------- WMMA GEMM: xout = relu((agg * nd) @ W + b) ----------------
// Block = 256 threads = 8 waves; W (64x64 f32 = 16 KB) staged once per block
// into LDS via gfx1250 async global->LDS copy, then each wave computes a
// 16-row x 64-col output panel with V_WMMA_F32_16X16X4_F32.
// A frag (16x4 f32): lane l -> A[l%16][k0 + 2*(l/16) + {0,1}]
// B frag (4x16 f32): lane l -> B[k0 + 2*(l/16) + {0,1}][l%16]
// C/D: vgpr r -> M = r + 8*(l/16), N = l%16

__global__ void gcn_gemm_relu(float* xout, const float* agg, const float* ndn,
                              const float* W, const float* bias, int n) {
    __shared__ float Wlds[DFEAT * DFEAT];

    // --- cooperative staging of W into LDS (all 256 threads, before any guard) ---
    {
        int t = threadIdx.x;                 // 0..255, each copies 4 float4s
        const float* gsrc = W + (size_t)t * 4;
        float* ldst = Wlds + t * 4;
#if __has_builtin(__builtin_amdgcn_global_load_async_to_lds_b128)
        __builtin_amdgcn_global_load_async_to_lds_b128(ASG_V4I(gsrc), ASL_V4I(ldst), 0, 0);
        __builtin_amdgcn_global_load_async_to_lds_b128(ASG_V4I(gsrc), ASL_V4I(ldst), 4096, 0);
        __builtin_amdgcn_global_load_async_to_lds_b128(ASG_V4I(gsrc), ASL_V4I(ldst), 8192, 0);
        __builtin_amdgcn_global_load_async_to_lds_b128(ASG_V4I(gsrc), ASL_V4I(ldst), 12288, 0);
#if __has_builtin(__builtin_amdgcn_s_wait_asynccnt)
        __builtin_amdgcn_s_wait_asynccnt(0);
#else
        asm volatile("s_wait_asynccnt 0" ::: "memory");
#endif
#else
        // fallback: synchronous stage
        for (int j = 0; j < 4; ++j)
            *(float4*)(ldst + j * 1024) = *(const float4*)(gsrc + j * 1024);
#endif
    }
    __syncthreads();

    int wave = threadIdx.x >> 5;
    int lane = threadIdx.x & 31;
    int tile = blockIdx.x * (blockDim.x >> 5) + wave;
    int row0 = tile * 16;
    if (row0 < n) {                        // wave-uniform: EXEC all-ones inside
        int half = lane >> 4;              // 0 or 1
        int l16  = lane & 15;
        int m    = row0 + l16;             // A row owned by this lane
        float nrm = ndn[m];

        // preload scaled A row-panel: 16 k-steps, 2 f32 each
        const float* arow = agg + (size_t)m * DFEAT + 2 * half;
        v2f afrag[16];
#pragma unroll
        for (int ks = 0; ks < 16; ++ks) {
            float2 t = *(const float2*)(arow + 4 * ks);
            afrag[ks].x = t.x * nrm;
            afrag[ks].y = t.y * nrm;
        }

#pragma unroll
        for (int n0 = 0; n0 < DFEAT; n0 += 16) {
            v8f acc = {};
#pragma unroll
            for (int ks = 0; ks < 16; ++ks) {
                int k = 4 * ks + 2 * half;
                v2f bfrag;
                bfrag.x = Wlds[(k + 0) * DFEAT + n0 + l16];
                bfrag.y = Wlds[(k + 1) * DFEAT + n0 + l16];
                acc = __builtin_amdgcn_wmma_f32_16x16x4_f32(
                    false, afrag[ks], false, bfrag, (short)0, acc, false, false);
            }
            float bv = bias[n0 + l16];
#pragma unroll
            for (int r = 0; r < 8; ++r) {
                int row = row0 + r + 8 * half;
                xout[(size_t)row * DFEAT + n0 + l16] = fmaxf(acc[r] + bv, 0.0f);
            }
        }
    }
}

// ---------------- final FC: out = fc_w @ relu(x) + fc_b ----------------

__global__ void gcn_fc_init(float* out, const float* fb) {
    int i = threadIdx.x;
    if (i < NOUT) out[i] = fb[i];
}

__global__ void gcn_fc_dot(float* out, const float* fw, const float* x, long nd) {
    __shared__ float red[256];
    int o = blockIdx.y;
    long start  = (long)blockIdx.x * blockDim.x + threadIdx.x;
    long stride = (long)gridDim.x * blockDim.x;
    const float* wr = fw + (size_t)o * nd;
    float acc = 0.0f;
    for (long i = start; i < nd; i += stride)
        acc += wr[i] * fmaxf(x[i], 0.0f);
    red[threadIdx.x] = acc;
    __syncthreads();
    for (int s = 128; s > 0; s >>= 1) {
        if ((int)threadIdx.x < s) red[threadIdx.x] += red[threadIdx.x + s];
        __syncthreads();
    }
    if (threadIdx.x == 0) atomicAdd(&out[o], red[0]);
}

// ---------------- launcher ----------------

extern "C" void kernel_launch(void* const* d_in, const int* in_sizes, int n_in,
                              void* d_out, int out_size, void* d_ws, size_t ws_size,
                              hipStream_t stream) {
    (void)n_in; (void)out_size; (void)ws_size;
    const float* F   = (const float*)d_in[0];
    const int*   src = (const int*)  d_in[1];
    const int*   dst = (const int*)  d_in[2];
    const float* gw  = (const float*)d_in[3];
    const float* gb  = (const float*)d_in[4];
    const float* fw  = (const float*)d_in[5];
    const float* fb  = (const float*)d_in[6];
    float* out = (float*)d_out;

    int n = in_sizes[0] / DFEAT;
    int e = in_sizes[1];
    long nd = (long)n * DFEAT;

    float* norm_src = (float*)d_ws;
    float* norm_dst = norm_src + n;
    float* x_ws     = norm_dst + n;
    float* agg      = x_ws + nd;

    gcn_init_deg<<<(n + 255) / 256, 256, 0, stream>>>(norm_src, norm_dst, n);
    gcn_count_deg<<<(e + 255) / 256, 256, 0, stream>>>(norm_src, norm_dst, src, dst, e);
    gcn_finish_norm<<<(n + 255) / 256, 256, 0, stream>>>(norm_src, norm_dst, n);

    int tiles = (n + 15) / 16;              // 16-row tiles, one per wave
    int blocks = (tiles + 7) / 8;           // 8 waves (256 threads) per block
    for (int layer = 0; layer < NLAYERS; ++layer) {
        const float* xin = (layer == 0) ? F : x_ws;
        gcn_init_agg<<<(int)((nd + 255) / 256), 256, 0, stream>>>(agg, xin, norm_src, nd);
        gcn_scatter<<<(int)(((long)e * 16 + 255) / 256), 256, 0, stream>>>(
            agg, xin, norm_src, src, dst, e);
        gcn_gemm_relu<<<blocks, 256, 0, stream>>>(
            x_ws, agg, norm_dst, gw + (size_t)layer * DFEAT * DFEAT, gb + layer * DFEAT, n);
    }

    gcn_fc_init<<<1, 32, 0, stream>>>(out, fb);
    dim3 grid(128, NOUT);
    gcn_fc_dot<<<grid, 256, 0, stream>>>(out, fw, x_ws, nd);
}